// HeteroGCNConv_58265526338121
// MI455X (gfx1250) — compile-verified
//
#include <hip/hip_runtime.h>

typedef __attribute__((ext_vector_type(2))) float v2f;
typedef __attribute__((ext_vector_type(8))) float v8f;

#define HDIM 128   // hidden size, fixed by the reference
#define NT   16    // WMMA tile

// ---------- utility kernels ----------

__global__ void fill_ones_kernel(float* __restrict__ p, int n) {
  int i = blockIdx.x * blockDim.x + threadIdx.x;
  if (i < n) p[i] = 1.0f;   // self-loop contributes 1 to both degrees
}

__global__ void zero_kernel(float* __restrict__ p, long long n) {
  long long i = (long long)blockIdx.x * blockDim.x + threadIdx.x;
  if (i < n) p[i] = 0.0f;
}

__global__ void degree_kernel(const int* __restrict__ src, const int* __restrict__ dst,
                              float* __restrict__ deg_out, float* __restrict__ deg_in, int E) {
  int e = blockIdx.x * blockDim.x + threadIdx.x;
  if (e < E) {
    atomicAdd(&deg_out[src[e]], 1.0f);
    atomicAdd(&deg_in[dst[e]], 1.0f);
  }
}

// in-place deg -> rsqrt(max(deg,1))
__global__ void norm_kernel(float* __restrict__ p, int n) {
  int i = blockIdx.x * blockDim.x + threadIdx.x;
  if (i < n) p[i] = rsqrtf(fmaxf(p[i], 1.0f));
}

// ---------- WMMA GEMM: Y[N x HDIM] = X[N x KD] * W[KD x HDIM] ----------
// block = 256 threads = 8 waves; wave w computes column tile w (8*16 = 128 cols)
// blockIdx.x = row tile. N, KD are multiples of 16 (50000/30000 rows, KD 256/128).
__global__ void gemm_wmma_f32_kernel(const float* __restrict__ X, const float* __restrict__ W,
                                     float* __restrict__ Y, int KD) {
  const int wave = threadIdx.x >> 5;
  const int lane = threadIdx.x & 31;
  const int l16  = lane & 15;
  const int hi   = lane >> 4;                 // 0: lanes 0-15, 1: lanes 16-31
  const int mbase = blockIdx.x * NT;
  const int nbase = wave * NT;

  const float* __restrict__ xrow = X + (size_t)(mbase + l16) * KD;
  const float* __restrict__ wcol = W + nbase + l16;

  v8f c = {};
  #pragma unroll 4
  for (int k = 0; k < KD; k += 4) {
    v2f a, b;
    // A 16x4 f32 fragment: VGPR0 = K(2*hi), VGPR1 = K(2*hi+1)
    a.x = xrow[k + 2 * hi];
    a.y = xrow[k + 2 * hi + 1];
    // B 4x16 f32 fragment: VGPR0 = row K(2*hi), VGPR1 = row K(2*hi+1)
    b.x = wcol[(size_t)(k + 2 * hi) * HDIM];
    b.y = wcol[(size_t)(k + 2 * hi + 1) * HDIM];
    c = __builtin_amdgcn_wmma_f32_16x16x4_f32(false, a, false, b, (short)0, c, false, false);
  }

  // D 16x16 f32: VGPR v -> row (v + 8*hi), col = lane&15
  float* __restrict__ yout = Y + nbase + l16;
  #pragma unroll
  for (int v = 0; v < 8; ++v)
    yout[(size_t)(mbase + v + 8 * hi) * HDIM] = c[v];
}

// ---------- edge scatter: agg[dst] += ns[src] * Y[src], one warp per edge ----------
__global__ void scatter_kernel(const int* __restrict__ src, const int* __restrict__ dst,
                               const float* __restrict__ ns, const float* __restrict__ Y,
                               float* __restrict__ agg, int E) {
  int warp = (int)(((long long)blockIdx.x * blockDim.x + threadIdx.x) >> 5);
  int lane = threadIdx.x & 31;
  if (warp >= E) return;
  int s = src[warp], d = dst[warp];
  float w = ns[s];
  float4 m = ((const float4*)(Y + (size_t)s * HDIM))[lane];
  float* ap = agg + (size_t)d * HDIM + lane * 4;
  atomicAdd(ap + 0, m.x * w);
  atomicAdd(ap + 1, m.y * w);
  atomicAdd(ap + 2, m.z * w);
  atomicAdd(ap + 3, m.w * w);
}

// ---------- finalize: out = ((agg + ns[i]*Y) * nd[i]) + b[j], optional relu ----------
__global__ void finalize_kernel(const float* __restrict__ agg, const float* __restrict__ Y,
                                const float* __restrict__ ns, const float* __restrict__ nd,
                                const float* __restrict__ bias, float* __restrict__ out,
                                long long total, int relu) {
  long long idx = (long long)blockIdx.x * blockDim.x + threadIdx.x;
  if (idx >= total) return;
  int i = (int)(idx >> 7);        // node
  int j = (int)(idx & (HDIM - 1)); // feature
  float v = (agg[idx] + Y[idx] * ns[i]) * nd[i] + bias[j];
  if (relu) v = fmaxf(v, 0.0f);
  out[idx] = v;
}

// ---------- host-side orchestration ----------
static inline int cdiv(long long a, int b) { return (int)((a + b - 1) / b); }

static void run_graph(const float* feat, const int* src, const int* dst,
                      const float* W0, const float* b0, const float* W1, const float* b1,
                      int N, int D, int E,
                      float* ns, float* nd, float* Y, float* agg, float* h,
                      float* out, hipStream_t stream) {
  const long long NH = (long long)N * HDIM;

  // degrees with self-loop baked in as the initial 1.0
  fill_ones_kernel<<<cdiv(2 * (long long)N, 256), 256, 0, stream>>>(ns, 2 * N); // ns,nd contiguous
  degree_kernel<<<cdiv(E, 256), 256, 0, stream>>>(src, dst, ns, nd, E);
  norm_kernel<<<cdiv(2 * (long long)N, 256), 256, 0, stream>>>(ns, 2 * N);

  // ---- layer 0: Y = feat @ W0 ; agg = scatter ; h = relu((agg + ns*Y)*nd + b0)
  gemm_wmma_f32_kernel<<<N / NT, 256, 0, stream>>>(feat, W0, Y, D);
  zero_kernel<<<cdiv(NH, 256), 256, 0, stream>>>(agg, NH);
  scatter_kernel<<<cdiv((long long)E * 32, 256), 256, 0, stream>>>(src, dst, ns, Y, agg, E);
  finalize_kernel<<<cdiv(NH, 256), 256, 0, stream>>>(agg, Y, ns, nd, b0, h, NH, 1);

  // ---- layer 1: Y = h @ W1 ; agg = scatter ; out = (agg + ns*Y)*nd + b1
  gemm_wmma_f32_kernel<<<N / NT, 256, 0, stream>>>(h, W1, Y, HDIM);
  zero_kernel<<<cdiv(NH, 256), 256, 0, stream>>>(agg, NH);
  scatter_kernel<<<cdiv((long long)E * 32, 256), 256, 0, stream>>>(src, dst, ns, Y, agg, E);
  finalize_kernel<<<cdiv(NH, 256), 256, 0, stream>>>(agg, Y, ns, nd, b1, out, NH, 0);
}

extern "C" void kernel_launch(void* const* d_in, const int* in_sizes, int n_in,
                              void* d_out, int out_size, void* d_ws, size_t ws_size,
                              hipStream_t stream) {
  const float* feat_a = (const float*)d_in[0];
  const float* feat_b = (const float*)d_in[1];
  const int*   src_a  = (const int*)d_in[2];
  const int*   dst_a  = (const int*)d_in[3];
  const int*   src_b  = (const int*)d_in[4];
  const int*   dst_b  = (const int*)d_in[5];
  const float* Wa0 = (const float*)d_in[6];
  const float* ba0 = (const float*)d_in[7];
  const float* Wa1 = (const float*)d_in[8];
  const float* ba1 = (const float*)d_in[9];
  const float* Wb0 = (const float*)d_in[10];
  const float* bb0 = (const float*)d_in[11];
  const float* Wb1 = (const float*)d_in[12];
  const float* bb1 = (const float*)d_in[13];

  const int DA = in_sizes[6] / HDIM;           // 256
  const int DB = in_sizes[10] / HDIM;          // 128
  const int NA = in_sizes[0] / DA;             // 50000
  const int NB = in_sizes[1] / DB;             // 30000
  const int EA = in_sizes[2];                  // 600000
  const int EB = in_sizes[4];                  // 300000

  // workspace layout (floats): per graph [ns | nd | Y | agg | h]
  float* ws = (float*)d_ws;
  float* ns_a  = ws;
  float* nd_a  = ns_a + NA;
  float* Y_a   = nd_a + NA;
  float* agg_a = Y_a + (size_t)NA * HDIM;
  float* h_a   = agg_a + (size_t)NA * HDIM;
  float* ns_b  = h_a + (size_t)NA * HDIM;
  float* nd_b  = ns_b + NB;
  float* Y_b   = nd_b + NB;
  float* agg_b = Y_b + (size_t)NB * HDIM;
  float* h_b   = agg_b + (size_t)NB * HDIM;

  float* out_a = (float*)d_out;
  float* out_b = out_a + (size_t)NA * HDIM;

  run_graph(feat_a, src_a, dst_a, Wa0, ba0, Wa1, ba1, NA, DA, EA,
            ns_a, nd_a, Y_a, agg_a, h_a, out_a, stream);
  run_graph(feat_b, src_b, dst_b, Wb0, bb0, Wb1, bb1, NB, DB, EB,
            ns_b, nd_b, Y_b, agg_b, h_b, out_b, stream);
}